// GAT_352187318573
// MI455X (gfx1250) — compile-verified
//
#include <hip/hip_runtime.h>
#include <hip/hip_bf16.h>

// ---------------------------------------------------------------------------
// GAT (2 layers) for MI455X / gfx1250, wave32.
//   per-layer:
//     s_src[n] = <h[n], aw[0:128]>, s_dst[n] = <h[n], aw[128:256]>
//     e[i]     = relu(s_src[src] + s_dst[dst] + ab)        (per edge)
//     softmax over incoming edges (segment max + sum via atomics)
//     z[d]    += alpha[i] * efeats[i]                      (atomic scatter)
//     hout     = relu([h|z] @ Ww^T + Wb)                   (WMMA f32 16x16x4)
// Bottleneck is efeats traffic (2 x 327 MB @ 23.3 TB/s) + L2 f32 atomics;
// the GEMM (2 x 3.3 GFLOP) runs in full f32 precision on the WMMA path.
// ---------------------------------------------------------------------------

typedef __attribute__((ext_vector_type(2))) float v2f;
typedef __attribute__((ext_vector_type(8))) float v8f;

#define WAVE 32

// ---------------- zero scratch ----------------
__global__ void zero_kernel(float* __restrict__ p, long long n) {
    long long i = (long long)blockIdx.x * blockDim.x + threadIdx.x;
    long long stride = (long long)gridDim.x * blockDim.x;
    for (; i < n; i += stride) p[i] = 0.0f;
}

// ---------------- pack B = Ww^T into WMMA-fragment layout ----------------
// Bt2[(k>>1)*256 + j*2 + (k&1)] = Ww[j*256 + k]   (Ww is [128,256] row major)
// => each lane's (b.x, b.y) = (B[k][j], B[k+1][j]) is one contiguous 8B load.
__global__ void pack_B(const float* __restrict__ Ww, float* __restrict__ Bt2) {
    int idx = blockIdx.x * blockDim.x + threadIdx.x;   // 0 .. 128*256-1
    if (idx >= 128 * 256) return;
    int j = idx >> 8;     // output feature 0..127
    int k = idx & 255;    // input feature  0..255
    Bt2[(k >> 1) * 256 + j * 2 + (k & 1)] = Ww[idx];
}

// ---------------- per-node attention partial dots (one wave per node) -------
__global__ void node_dots(const float* __restrict__ h, const float* __restrict__ aw,
                          float* __restrict__ s_src, float* __restrict__ s_dst, int N) {
    int wave = threadIdx.x >> 5, lane = threadIdx.x & 31;
    int n = blockIdx.x * (blockDim.x >> 5) + wave;
    if (n >= N) return;
    const float4 hx = ((const float4*)(h + (size_t)n * 128))[lane];
    const float4 w0 = ((const float4*)aw)[lane];          // aw[0:128]
    const float4 w1 = ((const float4*)(aw + 128))[lane];  // aw[128:256]
    float d0 = hx.x * w0.x + hx.y * w0.y + hx.z * w0.z + hx.w * w0.w;
    float d1 = hx.x * w1.x + hx.y * w1.y + hx.z * w1.z + hx.w * w1.w;
    #pragma unroll
    for (int off = 16; off > 0; off >>= 1) {
        d0 += __shfl_xor(d0, off, WAVE);
        d1 += __shfl_xor(d1, off, WAVE);
    }
    if (lane == 0) { s_src[n] = d0; s_dst[n] = d1; }
}

// ---------------- edge logits + segment max -------------------------------
// e >= 0 (relu) so integer atomicMax on the bit pattern == float max, and the
// 0-initialized emax is absorbed (segment max of relu'd values is >= 0).
__global__ void edge_logits(const int* __restrict__ src, const int* __restrict__ dst,
                            const float* __restrict__ s_src, const float* __restrict__ s_dst,
                            const float* __restrict__ ab,
                            float* __restrict__ e_ws, unsigned int* __restrict__ emax_u, int E) {
    int i = blockIdx.x * blockDim.x + threadIdx.x;
    if (i >= E) return;
    float e = s_src[src[i]] + s_dst[dst[i]] + ab[0];
    e = e > 0.0f ? e : 0.0f;
    e_ws[i] = e;
    atomicMax(&emax_u[dst[i]], __float_as_uint(e));
}

// ---------------- exp + segment sum ---------------------------------------
__global__ void edge_exp(const int* __restrict__ dst, const float* __restrict__ e_ws,
                         const float* __restrict__ emax,
                         float* __restrict__ ex_ws, float* __restrict__ denom, int E) {
    int i = blockIdx.x * blockDim.x + threadIdx.x;
    if (i >= E) return;
    int d = dst[i];
    float ex = __expf(e_ws[i] - emax[d]);
    ex_ws[i] = ex;
    unsafeAtomicAdd(&denom[d], ex);  // native global_atomic_add_f32
}

// ---------------- weighted scatter of edge features (one wave per edge) ----
__global__ void edge_scatter(const int* __restrict__ dst, const float* __restrict__ ex_ws,
                             const float* __restrict__ denom, const float* __restrict__ efeats,
                             float* __restrict__ z, int E) {
    int wave = threadIdx.x >> 5, lane = threadIdx.x & 31;
    int i = blockIdx.x * (blockDim.x >> 5) + wave;
    if (i >= E) return;
    int d = dst[i];
    float alpha = ex_ws[i] / denom[d];
    float4 m = ((const float4*)(efeats + (size_t)i * 128))[lane];
    float* zp = z + (size_t)d * 128 + lane * 4;
    unsafeAtomicAdd(zp + 0, alpha * m.x);
    unsafeAtomicAdd(zp + 1, alpha * m.y);
    unsafeAtomicAdd(zp + 2, alpha * m.z);
    unsafeAtomicAdd(zp + 3, alpha * m.w);
}

// ---------------- node update GEMM with V_WMMA_F32_16X16X4_F32 -------------
// out[16 nodes x 128 feats] per wave; A = [h | z] (M=16,K=256), B = Ww^T.
// A-frag (16x4 f32): lane L holds row M=L%16; lanes 0-15 carry K={k0,k0+1},
// lanes 16-31 carry K={k0+2,k0+3} (2 VGPRs). B-frag symmetric in N.
// K loop is split at the h/z boundary (K=128) so each half has one
// loop-invariant A base pointer -> pure base+offset addressing, no cndmask.
// Grid is exact (3125 tiles = 625 blocks x 5 waves) -> EXEC all-1s, no guard.
__device__ __forceinline__ void gemm_khalf(const float* __restrict__ Abase,
                                           const float* __restrict__ Bhalf,
                                           int col, v8f acc[8]) {
    #pragma unroll 4
    for (int k0 = 0; k0 < 128; k0 += 4) {
        v2f a = *(const v2f*)(Abase + k0);                    // A[row][k..k+1]
        const float* Bbase = Bhalf + (size_t)(k0 >> 1) * 256;
        #pragma unroll
        for (int t = 0; t < 8; ++t) {
            v2f b = *(const v2f*)(Bbase + (t * 16 + col) * 2); // B[k][j],B[k+1][j]
            acc[t] = __builtin_amdgcn_wmma_f32_16x16x4_f32(
                false, a, false, b, (short)0, acc[t], false, false);
        }
    }
}

__global__ void __launch_bounds__(160)
node_update_wmma(const float* __restrict__ hin, const float* __restrict__ z,
                 const float* __restrict__ Bt2, const float* __restrict__ bias,
                 float* __restrict__ hout) {
    const int wave = threadIdx.x >> 5;
    const int lane = threadIdx.x & 31;
    const int tile = blockIdx.x * 5 + wave;       // 0 .. 3124
    const int col  = lane & 15;
    const int koff = (lane >> 4) << 1;            // 0 for lanes 0-15, 2 for 16-31
    const int row  = tile * 16 + col;             // A row this lane feeds

    v8f acc[8] = {};                              // 8 N-tiles of 16 -> 128 outputs

    // K half 0: A = hin, B rows 0..127 ; K half 1: A = z, B rows 128..255
    gemm_khalf(hin + (size_t)row * 128 + koff, Bt2 + (size_t)(koff >> 1) * 256,
               col, acc);
    gemm_khalf(z + (size_t)row * 128 + koff,
               Bt2 + (size_t)((128 + koff) >> 1) * 256, col, acc);

    // D layout: VGPR v, lanes 0-15 -> M=v, lanes 16-31 -> M=v+8, N=lane%16
    const int mbase = tile * 16 + ((lane >> 4) << 3);
    #pragma unroll
    for (int t = 0; t < 8; ++t) {
        const int j = t * 16 + col;
        const float bj = bias[j];
        #pragma unroll
        for (int v = 0; v < 8; ++v) {
            float val = acc[t][v] + bj;
            hout[(size_t)(mbase + v) * 128 + j] = val > 0.0f ? val : 0.0f;
        }
    }
}

// ---------------------------------------------------------------------------
static void run_layer(const float* hin, const float* efeats, const float* Bt2,
                      const float* Wb, const float* aw, const float* ab,
                      const int* src, const int* dst,
                      float* s_src, float* s_dst, float* emax, float* denom,
                      float* zbuf, float* e_ws, float* ex_ws, float* hout,
                      int N, int E, hipStream_t stream) {
    // emax, denom, z are contiguous in the workspace: one zeroing pass
    long long nz = (long long)N * 2 + (long long)N * 128;
    hipLaunchKernelGGL(zero_kernel, dim3(2048), dim3(256), 0, stream, emax, nz);
    hipLaunchKernelGGL(node_dots, dim3((N + 7) / 8), dim3(256), 0, stream,
                       hin, aw, s_src, s_dst, N);
    hipLaunchKernelGGL(edge_logits, dim3((E + 255) / 256), dim3(256), 0, stream,
                       src, dst, s_src, s_dst, ab, e_ws, (unsigned int*)emax, E);
    hipLaunchKernelGGL(edge_exp, dim3((E + 255) / 256), dim3(256), 0, stream,
                       dst, e_ws, emax, ex_ws, denom, E);
    hipLaunchKernelGGL(edge_scatter, dim3((E + 7) / 8), dim3(256), 0, stream,
                       dst, ex_ws, denom, efeats, zbuf, E);
    int nTiles = N / 16;                          // 3125 = 625 * 5, exact
    hipLaunchKernelGGL(node_update_wmma, dim3(nTiles / 5), dim3(160), 0, stream,
                       hin, zbuf, Bt2, Wb, hout);
}

extern "C" void kernel_launch(void* const* d_in, const int* in_sizes, int n_in,
                              void* d_out, int out_size, void* d_ws, size_t ws_size,
                              hipStream_t stream) {
    const float* nfeats = (const float*)d_in[0];
    const float* efeats = (const float*)d_in[1];
    const float* Ww0    = (const float*)d_in[2];
    const float* Wb0    = (const float*)d_in[3];
    const float* aw0    = (const float*)d_in[4];
    const float* ab0    = (const float*)d_in[5];
    const float* Ww1    = (const float*)d_in[6];
    const float* Wb1    = (const float*)d_in[7];
    const float* aw1    = (const float*)d_in[8];
    const float* ab1    = (const float*)d_in[9];
    const int*   src    = (const int*)d_in[10];
    const int*   dst    = (const int*)d_in[11];

    const int N = in_sizes[0] / 128;   // 50000
    const int E = in_sizes[10];        // 640000

    // workspace layout (floats)
    float* ws    = (float*)d_ws;
    float* h1    = ws;  ws += (size_t)N * 128;
    float* s_src = ws;  ws += N;
    float* s_dst = ws;  ws += N;
    float* emax  = ws;  ws += N;        // |
    float* denom = ws;  ws += N;        // | contiguous: zeroed together
    float* zbuf  = ws;  ws += (size_t)N * 128;   // |
    float* e_ws  = ws;  ws += E;
    float* ex_ws = ws;  ws += E;
    float* Bt0   = ws;  ws += 128 * 256;
    float* Bt1   = ws;  ws += 128 * 256;

    hipLaunchKernelGGL(pack_B, dim3(128), dim3(256), 0, stream, Ww0, Bt0);
    hipLaunchKernelGGL(pack_B, dim3(128), dim3(256), 0, stream, Ww1, Bt1);

    run_layer(nfeats, efeats, Bt0, Wb0, aw0, ab0, src, dst,
              s_src, s_dst, emax, denom, zbuf, e_ws, ex_ws, h1, N, E, stream);
    run_layer(h1, efeats, Bt1, Wb1, aw1, ab1, src, dst,
              s_src, s_dst, emax, denom, zbuf, e_ws, ex_ws, (float*)d_out,
              N, E, stream);
}